// MultiPf_11175504904264
// MI455X (gfx1250) — compile-verified
//
#include <hip/hip_runtime.h>
#include <math.h>

// Problem constants (match reference)
#define NPFS 16
#define NF   2048
#define NE   32
#define BATCH 2048

typedef __attribute__((ext_vector_type(2))) float v2f;
typedef __attribute__((ext_vector_type(8))) float v8f;

// D(16x16) = A(16x4) x B(4x16) + C, f32 WMMA (wave32).
// 8-arg pattern: (neg_a, A, neg_b, B, c_mod, C, reuse_a, reuse_b)
__device__ __forceinline__ v8f wmma_f32_k4(v2f a, v2f b, v8f c) {
    return __builtin_amdgcn_wmma_f32_16x16x4_f32(false, a, false, b, (short)0, c,
                                                 false, false);
}

__device__ __forceinline__ float bcast_lane(float x, int lane) {
    return __int_as_float(__builtin_amdgcn_readlane(__float_as_int(x), lane));
}

// One block per batch b (512 threads = 16 waves); wave w handles matrix f = w.
// 32x32 antisymmetric matrix held as 2x2 grid of 16x16 f32 WMMA C/D tiles:
//   tile(bi,bj) reg r, lane l: row i = 16*bi + r + 8*(l>>4), col j = 16*bj + (l&15)
__global__ __launch_bounds__(512)
void multi_pf_kernel(const float* __restrict__ F, const int* __restrict__ gidx,
                     float* __restrict__ out) {
    const int b      = blockIdx.x;
    const int lane   = threadIdx.x & 31;
    const int wave   = threadIdx.x >> 5;   // = f in [0,16)
    const int lane16 = lane & 15;
    const int h      = lane >> 4;          // 0 = rows r, 1 = rows r+8 within tile
    const bool lh    = (lane < 16);

    __shared__ int   sidx[NE];
    __shared__ float ssig[NPFS];
    __shared__ float slog[NPFS];

    if (threadIdx.x < NE) sidx[threadIdx.x] = gidx[(size_t)b * NE + threadIdx.x];
    __syncthreads();

    // ---- Gather 32x32 submatrix and antisymmetrize: A = 0.5*(G - G^T) ----
    const float* Fo = F + (size_t)wave * (size_t)NF * (size_t)NF;
    const int cj0 = sidx[lane16];
    const int cj1 = sidx[16 + lane16];

    v8f Cm[2][2];
    #pragma unroll
    for (int bi = 0; bi < 2; ++bi) {
        #pragma unroll
        for (int r = 0; r < 8; ++r) {
            const int i  = 16 * bi + r + 8 * h;
            const int ri = sidx[i];
            {
                float g  = Fo[(size_t)ri  * NF + cj0];
                float gt = Fo[(size_t)cj0 * NF + ri ];
                Cm[bi][0][r] = 0.5f * (g - gt);
            }
            {
                float g  = Fo[(size_t)ri  * NF + cj1];
                float gt = Fo[(size_t)cj1 * NF + ri ];
                Cm[bi][1][r] = 0.5f * (g - gt);
            }
        }
    }

    // ---- Parlett-Reid elimination: 15 steps, rank-2 update via WMMA ----
    // log|pf| tracked exactly as (product of pivot mantissas, sum of exponents);
    // mantissas in [0.5,1) -> product of 16 >= 1.5e-5, no underflow. One logf at end.
    float sgn   = 1.0f;
    float mprod = 1.0f;
    int   esum  = 0;

    #pragma unroll
    for (int s = 0; s < 15; ++s) {
        const int k  = 2 * s;
        // Extract row k as a lane-vector: rowk[l] = A[k][l]
        const int bk = k >> 4;
        const int rg = k & 7;                 // (k&15)&7, k even
        const int sb = (k & 8) ? 16 : 0;
        float lo   = __shfl(Cm[bk][0][rg], sb + lane16, 32);
        float hi   = __shfl(Cm[bk][1][rg], sb + lane16, 32);
        float rowk = lh ? lo : hi;
        // Row k+1 (same 16-block since k even)
        const int rg1 = (k + 1) & 7;
        const int sb1 = ((k + 1) & 8) ? 16 : 0;
        float lo1   = __shfl(Cm[bk][0][rg1], sb1 + lane16, 32);
        float hi1   = __shfl(Cm[bk][1][rg1], sb1 + lane16, 32);
        float rowk1 = lh ? lo1 : hi1;

        // Pivot p = A[k][k+1], wave-uniform via v_readlane
        float p = bcast_lane(rowk, k + 1);
        sgn *= (p > 0.0f) ? 1.0f : ((p < 0.0f) ? -1.0f : 0.0f);
        int e; float mant = frexpf(fabsf(p), &e);
        mprod *= mant; esum += e;

        // tau = row_k / p (uniform reciprocal), v = row_{k+1}; zero below k+2 so the
        // full 32x32 rank-2 update equals the reference's shrinking-submatrix update.
        float rp  = __builtin_amdgcn_rcpf(p);
        float tau = (lane >= k + 2) ? rowk * rp : 0.0f;
        float vv  = (lane >= k + 2) ? rowk1     : 0.0f;

        // Upper-half slices (cols/rows 16..31) brought into lanes 0-15.
        float t_hi = __shfl(tau, 16 + lane16, 32);
        float v_hi = __shfl(vv,  16 + lane16, 32);

        // A(16x4) = [v | tau | 0 | 0]; lanes 0-15 carry K0,K1, lanes 16-31 = 0.
        // Lanes 0-15 of tau/vv already hold the low-slice values directly.
        v2f A0 = { lh ? vv   : 0.0f, lh ? tau  : 0.0f };
        v2f A1 = { lh ? v_hi : 0.0f, lh ? t_hi : 0.0f };
        // B(4x16) rows = [tau ; -v ; 0 ; 0]: reg0 = {K0|K2}, reg1 = {K1|K3}
        v2f B0 = { lh ? tau  : 0.0f, lh ? -vv   : 0.0f };
        v2f B1 = { lh ? t_hi : 0.0f, lh ? -v_hi : 0.0f };

        // D = v (x) tau - tau (x) v + C on the four 16x16 tiles (dead ones DCE'd)
        Cm[0][0] = wmma_f32_k4(A0, B0, Cm[0][0]);
        Cm[0][1] = wmma_f32_k4(A0, B1, Cm[0][1]);
        Cm[1][0] = wmma_f32_k4(A1, B0, Cm[1][0]);
        Cm[1][1] = wmma_f32_k4(A1, B1, Cm[1][1]);
    }

    // Final pivot A[30][31]: tile(1,1), in-tile row 14 -> reg 6 upper half, col 15 -> lane 31
    {
        float pf = bcast_lane(Cm[1][1][6], 31);
        sgn *= (pf > 0.0f) ? 1.0f : ((pf < 0.0f) ? -1.0f : 0.0f);
        int e; float mant = frexpf(fabsf(pf), &e);
        mprod *= mant; esum += e;
    }

    const float lga = logf(mprod) + (float)esum * 0.6931471805599453f;

    if (lane == 0) { ssig[wave] = sgn; slog[wave] = lga; }
    __syncthreads();

    // ---- Signed logsumexp over the 16 matrices ----
    if (threadIdx.x == 0) {
        float m = -INFINITY;
        #pragma unroll
        for (int f2 = 0; f2 < NPFS; ++f2) m = fmaxf(m, slog[f2]);
        float val = 0.0f;
        #pragma unroll
        for (int f2 = 0; f2 < NPFS; ++f2) val += ssig[f2] * expf(slog[f2] - m);
        float osgn = (val > 0.0f) ? 1.0f : ((val < 0.0f) ? -1.0f : 0.0f);
        out[b]         = osgn;                   // out_sign
        out[BATCH + b] = m + logf(fabsf(val));   // out_logabs
    }
}

extern "C" void kernel_launch(void* const* d_in, const int* in_sizes, int n_in,
                              void* d_out, int out_size, void* d_ws, size_t ws_size,
                              hipStream_t stream) {
    const float* F   = (const float*)d_in[0];   // (16, 2048, 2048) f32
    const int*   idx = (const int*)d_in[1];     // (2048, 32) i32
    float* out = (float*)d_out;                 // [sign(2048) | logabs(2048)]
    (void)in_sizes; (void)n_in; (void)out_size; (void)d_ws; (void)ws_size;

    multi_pf_kernel<<<BATCH, 512, 0, stream>>>(F, idx, out);
}